// CustomYOLOModel_68771016343746
// MI455X (gfx1250) — compile-verified
//
#include <hip/hip_runtime.h>
#include <hip/hip_bf16.h>
#include <stdint.h>

#define NA    134400      // anchors
#define NC    365         // classes
#define KDET  300
#define NCHUNK 4
#define CPC   92          // classes per chunk (last chunk has 89)
#define ABLK  525         // anchor blocks (256 anchors each)

// ---------------- workspace layout (bytes) ----------------
// scores  f32[NA]            @ 0
// cid     u32[NA]            @ 537600
// pscore  f32[4*NA]          @ 1075200
// pcid    u32[4*NA]          @ 3225600
// hist    u32[4*65536]       @ 5376000
// state   {u64 prefix; u32 k; u32 counter} @ 6424576
// keyl    u64[300]           @ 6424592
// idxl    u32[300]           @ 6426992
struct SelState { unsigned long long prefix; unsigned k; unsigned counter; };

static const size_t OFF_SCORES = 0;
static const size_t OFF_CID    = 537600;
static const size_t OFF_PSC    = 1075200;
static const size_t OFF_PCI    = 3225600;
static const size_t OFF_HIST   = 5376000;
static const size_t OFF_STATE  = 6424576;
static const size_t OFF_KEYL   = 6424592;
static const size_t OFF_IDXL   = 6426992;

// ---------------- CDNA5 async copy helpers ----------------
__device__ __forceinline__ unsigned lds_u32(const void* p) {
    // low 32 bits of a generic pointer into LDS == LDS byte offset
    return (unsigned)(uintptr_t)p;
}

__device__ __forceinline__ void async_copy16(unsigned lds_byte_off,
                                             const float* sbase,
                                             unsigned gvoff_bytes) {
    // GVS mode: addr = SGPR64 + VGPR32 ; 16 bytes -> LDS, tracked by ASYNCcnt
    asm volatile("global_load_async_to_lds_b128 %0, %1, %2"
                 :: "v"(lds_byte_off), "v"(gvoff_bytes), "s"(sbase)
                 : "memory");
}

__device__ __forceinline__ void wait_async3() {
    asm volatile("s_wait_asynccnt 3" ::: "memory");
}
__device__ __forceinline__ void wait_async0() {
    asm volatile("s_wait_asynccnt 0" ::: "memory");
}
__device__ __forceinline__ void wait_ds0() {
    asm volatile("s_wait_dscnt 0" ::: "memory");
}

__device__ __forceinline__ unsigned mono_key32(float f) {
    unsigned u = __float_as_uint(f);
    return (u & 0x80000000u) ? ~u : (u | 0x80000000u);
}

// ---------------- K0: init histograms + state ----------------
__global__ void k_init(unsigned* __restrict__ hist, SelState* __restrict__ st) {
    int i = blockIdx.x * blockDim.x + threadIdx.x;
    if (i < 4 * 65536) hist[i] = 0u;
    if (i == 0) { st->prefix = 0ull; st->k = KDET; st->counter = 0u; }
}

// ---------------- K1: per-chunk class max/argmax via async LDS pipeline ----
__global__ void __launch_bounds__(64)
k_phase1(const float* __restrict__ in, float* __restrict__ pscore,
         unsigned* __restrict__ pcid) {
    __shared__ float4 buf[4][64];
    const int t      = threadIdx.x;
    const int blk    = blockIdx.x;
    const int chunk  = blk / ABLK;
    const int a0     = (blk % ABLK) * 256 + t * 4;   // this thread's 4 anchors
    const int c0     = chunk * CPC;
    const int n      = min(NC, c0 + CPC) - c0;
    const unsigned gvoff = (unsigned)a0 * 4u;

    // prefetch depth 3
    #pragma unroll
    for (int s = 0; s < 3; ++s) {
        const float* base = in + (size_t)(4 + c0 + s) * NA;
        async_copy16(lds_u32(&buf[s][t]), base, gvoff);
    }

    float best0 = -1e30f, best1 = -1e30f, best2 = -1e30f, best3 = -1e30f;
    unsigned bid0 = 0, bid1 = 0, bid2 = 0, bid3 = 0;

    for (int c = 0; c < n; ++c) {
        wait_ds0();                        // last read of reused slot finished
        if (c + 3 < n) {
            const float* base = in + (size_t)(4 + c0 + c + 3) * NA;
            async_copy16(lds_u32(&buf[(c + 3) & 3][t]), base, gvoff);
            wait_async3();                 // oldest (slot c) landed in LDS
        } else {
            wait_async0();                 // pipeline drain at tail
        }
        float4 v = buf[c & 3][t];
        unsigned cc = (unsigned)(c0 + c);
        if (v.x > best0) { best0 = v.x; bid0 = cc; }
        if (v.y > best1) { best1 = v.y; bid1 = cc; }
        if (v.z > best2) { best2 = v.z; bid2 = cc; }
        if (v.w > best3) { best3 = v.w; bid3 = cc; }
    }

    size_t o = (size_t)chunk * NA + (size_t)a0;
    *(float4*)(pscore + o) = make_float4(best0, best1, best2, best3);
    *(uint4*)(pcid + o)    = make_uint4(bid0, bid1, bid2, bid3);
}

// ---------------- K2: combine chunks, mask, radix hist round 0 ------------
__global__ void k_combine(const float* __restrict__ pscore,
                          const unsigned* __restrict__ pcid,
                          float* __restrict__ scores, unsigned* __restrict__ cid,
                          unsigned* __restrict__ hist0) {
    int i = blockIdx.x * blockDim.x + threadIdx.x;        // exactly NA threads
    float best = pscore[i];
    unsigned bc = pcid[i];
    #pragma unroll
    for (int ch = 1; ch < NCHUNK; ++ch) {
        float v = pscore[(size_t)ch * NA + i];
        unsigned c = pcid[(size_t)ch * NA + i];
        if (v > best) { best = v; bc = c; }               // strict > keeps lowest class on ties
    }
    float masked = (best >= 0.25f) ? best : -1.0f;
    scores[i] = masked;
    cid[i] = bc;
    unsigned m = mono_key32(masked);
    atomicAdd(&hist0[m >> 16], 1u);                       // digit = key bits [63:48]
}

// ---------------- K3: radix hist rounds 1..3 ------------------------------
__global__ void k_hist(const float* __restrict__ scores,
                       const SelState* __restrict__ st,
                       unsigned* __restrict__ hist, int r) {
    int i = blockIdx.x * blockDim.x + threadIdx.x;
    unsigned m = mono_key32(scores[i]);
    unsigned long long key = ((unsigned long long)m << 32) | (unsigned)(~(unsigned)i);
    int shift = 64 - 16 * r;
    if ((key >> shift) == st->prefix) {
        unsigned d = (unsigned)((key >> (shift - 16)) & 0xFFFFull);
        atomicAdd(&hist[d], 1u);
    }
}

// ---------------- K4: select the digit containing the k-th largest --------
__global__ void k_select(const unsigned* __restrict__ hist, SelState* __restrict__ st) {
    __shared__ unsigned part[256];
    int t = threadIdx.x;
    unsigned s = 0;
    for (int j = 0; j < 256; ++j) s += hist[t * 256 + j];
    part[t] = s;
    __syncthreads();
    if (t == 0) {
        unsigned k = st->k;
        unsigned acc = 0;            // count of keys with digit strictly greater
        int seg = 255;
        for (; seg > 0; --seg) {
            if (acc + part[seg] >= k) break;
            acc += part[seg];
        }
        int d = seg * 256 + 255;
        for (; d > seg * 256; --d) {
            unsigned h = hist[d];
            if (acc + h >= k) break;
            acc += h;
        }
        // land on d even if the inner loop fell through to seg*256
        st->prefix = (st->prefix << 16) | (unsigned)d;
        st->k = k - acc;
    }
}

// ---------------- K5: compact the exact top-300 ---------------------------
__global__ void k_compact(const float* __restrict__ scores, SelState* __restrict__ st,
                          unsigned long long* __restrict__ keyl,
                          unsigned* __restrict__ idxl) {
    int i = blockIdx.x * blockDim.x + threadIdx.x;
    unsigned m = mono_key32(scores[i]);
    unsigned long long key = ((unsigned long long)m << 32) | (unsigned)(~(unsigned)i);
    if (key >= st->prefix) {                               // prefix == full 64-bit threshold
        unsigned p = atomicAdd(&st->counter, 1u);
        if (p < KDET) { keyl[p] = key; idxl[p] = (unsigned)i; }
    }
}

// ---------------- K6: sort, decode boxes, NMS, write outputs --------------
__global__ void __launch_bounds__(512)
k_final(const float* __restrict__ in, const float* __restrict__ scores,
        const unsigned* __restrict__ cid,
        const unsigned long long* __restrict__ keyl,
        const unsigned* __restrict__ idxl, float* __restrict__ out) {
    __shared__ unsigned long long sk[512];
    __shared__ unsigned si[512];
    __shared__ float bx[KDET], by[KDET], bw[KDET], bh[KDET], ssc[KDET];
    __shared__ unsigned smat[KDET * 10];
    __shared__ unsigned kw[10];
    int t = threadIdx.x;

    sk[t] = (t < KDET) ? keyl[t] : 0ull;   // 0 < any real key (mono map > 0)
    si[t] = (t < KDET) ? idxl[t] : 0u;
    __syncthreads();

    // bitonic sort, descending by key (=> desc score, asc index on ties)
    for (int k = 2; k <= 512; k <<= 1) {
        for (int j = k >> 1; j > 0; j >>= 1) {
            int ixj = t ^ j;
            if (ixj > t) {
                unsigned long long a = sk[t], b = sk[ixj];
                bool up = ((t & k) == 0);
                if (up ? (a < b) : (a > b)) {
                    sk[t] = b; sk[ixj] = a;
                    unsigned tmp = si[t]; si[t] = si[ixj]; si[ixj] = tmp;
                }
            }
            __syncthreads();
        }
    }

    if (t < KDET) {
        unsigned a = si[t];
        float x = in[a], y = in[NA + a], w = in[2 * NA + a], h = in[3 * NA + a];
        float left = truncf(x - 0.5f * w);
        float top  = truncf(y - 0.5f * h);
        float ww   = truncf(w);
        float hh   = truncf(h);
        bx[t] = left; by[t] = top; bw[t] = ww; bh[t] = hh;
        float sc = scores[a];
        ssc[t] = sc;
        out[4 * t + 0] = left;
        out[4 * t + 1] = top;
        out[4 * t + 2] = ww;
        out[4 * t + 3] = hh;
        out[1200 + t] = sc;
        out[1500 + t] = (float)cid[a];
    }
    if (t < 10) kw[t] = 0u;
    __syncthreads();

    // 300x300 suppression bit-matrix: smat[i][w] bit b => suppress j=w*32+b (j>i, iou>0.5)
    for (int task = t; task < KDET * 10; task += 512) {
        int i = task / 10, w = task % 10;
        float ix = bx[i], iy = by[i], iw = bw[i], ih = bh[i];
        float iArea = iw * ih;
        unsigned bits = 0;
        #pragma unroll 4
        for (int b = 0; b < 32; ++b) {
            int j = w * 32 + b;
            if (j < KDET && j > i) {
                float x1 = fmaxf(ix, bx[j]);
                float y1 = fmaxf(iy, by[j]);
                float x2 = fminf(ix + iw, bx[j] + bw[j]);
                float y2 = fminf(iy + ih, by[j] + bh[j]);
                float inter = fmaxf(x2 - x1, 0.0f) * fmaxf(y2 - y1, 0.0f);
                float uni = iArea + bw[j] * bh[j] - inter;
                float iou = inter / fmaxf(uni, 1e-9f);
                if (iou > 0.5f) bits |= (1u << b);
            }
        }
        smat[task] = bits;
    }
    __syncthreads();
    if (t < KDET && ssc[t] >= 0.25f) atomicOr(&kw[t >> 5], 1u << (t & 31));
    __syncthreads();

    if (t == 0) {
        unsigned K[10];
        #pragma unroll
        for (int w = 0; w < 10; ++w) K[w] = kw[w];
        for (int i = 0; i < KDET; ++i) {
            if ((K[i >> 5] >> (i & 31)) & 1u) {
                #pragma unroll
                for (int w = 0; w < 10; ++w) K[w] &= ~smat[i * 10 + w];
            }
        }
        #pragma unroll
        for (int w = 0; w < 10; ++w) kw[w] = K[w];
    }
    __syncthreads();
    if (t < KDET) out[1800 + t] = ((kw[t >> 5] >> (t & 31)) & 1u) ? 1.0f : 0.0f;
}

// ---------------- launcher ----------------
extern "C" void kernel_launch(void* const* d_in, const int* in_sizes, int n_in,
                              void* d_out, int out_size, void* d_ws, size_t ws_size,
                              hipStream_t stream) {
    (void)in_sizes; (void)n_in; (void)out_size; (void)ws_size;
    const float* in = (const float*)d_in[0];
    float* out = (float*)d_out;
    char* ws = (char*)d_ws;

    float*               scores = (float*)(ws + OFF_SCORES);
    unsigned*            cid    = (unsigned*)(ws + OFF_CID);
    float*               pscore = (float*)(ws + OFF_PSC);
    unsigned*            pcid   = (unsigned*)(ws + OFF_PCI);
    unsigned*            hist   = (unsigned*)(ws + OFF_HIST);
    SelState*            st     = (SelState*)(ws + OFF_STATE);
    unsigned long long*  keyl   = (unsigned long long*)(ws + OFF_KEYL);
    unsigned*            idxl   = (unsigned*)(ws + OFF_IDXL);

    k_init<<<1024, 256, 0, stream>>>(hist, st);
    k_phase1<<<ABLK * NCHUNK, 64, 0, stream>>>(in, pscore, pcid);
    k_combine<<<ABLK, 256, 0, stream>>>(pscore, pcid, scores, cid, hist);
    k_select<<<1, 256, 0, stream>>>(hist, st);
    for (int r = 1; r < 4; ++r) {
        k_hist<<<ABLK, 256, 0, stream>>>(scores, st, hist + (size_t)r * 65536, r);
        k_select<<<1, 256, 0, stream>>>(hist + (size_t)r * 65536, st);
    }
    k_compact<<<ABLK, 256, 0, stream>>>(scores, st, keyl, idxl);
    k_final<<<1, 512, 0, stream>>>(in, scores, cid, keyl, idxl, out);
}